// Decoder_3882650436292
// MI455X (gfx1250) — compile-verified
//
#include <hip/hip_runtime.h>
#include <hip/hip_bf16.h>
#include <math.h>

// Problem constants (match reference)
#define V_   50000
#define EMB_ 128
#define H_   256
#define TK_  400
#define B_   128
#define OOV_ 10
#define N2_  512       // 2H
#define VP1_ 50010     // V + OOV

#define NTILES_ (V_ / 16)   // 3125
#define NT_ 4               // n-tiles per wave
#define MT_ 4               // m-tiles per wave (M blocked 128 = 2 * 64)

typedef float v2f __attribute__((ext_vector_type(2)));
typedef float v8f __attribute__((ext_vector_type(8)));

// ---------------------------------------------------------------------------
// K1: x = concat(c_t_1, embedding[y_t_1]) @ Wx.T + bx    -> x [B,128]
// grid(B), block(128)
__global__ void k_x(const float* __restrict__ c_t_1, const int* __restrict__ y,
                    const float* __restrict__ emb, const float* __restrict__ Wx,
                    const float* __restrict__ bx, float* __restrict__ x) {
    int b = blockIdx.x, e = threadIdx.x;
    const float* row = Wx + e * 640;
    const float* cb  = c_t_1 + b * N2_;
    const float* ye  = emb + (long)y[b] * EMB_;
    float acc = bx[e];
    #pragma unroll 4
    for (int k = 0; k < N2_; ++k) acc += cb[k] * row[k];
    #pragma unroll 4
    for (int k = 0; k < EMB_; ++k) acc += ye[k] * row[N2_ + k];
    x[b * EMB_ + e] = acc;
}

// ---------------------------------------------------------------------------
// K2: single-step LSTM (gate order i,f,g,o) -> s_t_hat=[h1,c1], h1, c1
// grid(B), block(256)
__global__ void k_lstm(const float* __restrict__ x, const float* __restrict__ h0,
                       const float* __restrict__ c0,
                       const float* __restrict__ w_ih, const float* __restrict__ b_ih,
                       const float* __restrict__ w_hh, const float* __restrict__ b_hh,
                       float* __restrict__ s_t_hat,
                       float* __restrict__ h1_out, float* __restrict__ c1_out) {
    int b = blockIdx.x, h = threadIdx.x;
    const float* xb = x  + b * EMB_;
    const float* hb = h0 + b * H_;
    float g4[4];
    #pragma unroll
    for (int gi = 0; gi < 4; ++gi) {
        int j = gi * H_ + h;
        float acc = b_ih[j] + b_hh[j];
        const float* wi = w_ih + (long)j * EMB_;
        #pragma unroll 4
        for (int k = 0; k < EMB_; ++k) acc += xb[k] * wi[k];
        const float* wh = w_hh + (long)j * H_;
        #pragma unroll 4
        for (int k = 0; k < H_; ++k) acc += hb[k] * wh[k];
        g4[gi] = acc;
    }
    float ig = 1.f / (1.f + expf(-g4[0]));
    float fg = 1.f / (1.f + expf(-g4[1]));
    float gg = tanhf(g4[2]);
    float og = 1.f / (1.f + expf(-g4[3]));
    float c1 = fg * c0[b * H_ + h] + ig * gg;
    float h1 = og * tanhf(c1);
    s_t_hat[b * N2_ + h]       = h1;
    s_t_hat[b * N2_ + H_ + h]  = c1;
    h1_out[b * H_ + h] = h1;
    c1_out[b * H_ + h] = c1;
}

// ---------------------------------------------------------------------------
// K3: dec_fea = s_t_hat @ Wp.T + bp    [B,512]
// grid(B), block(512)
__global__ void k_decfea(const float* __restrict__ s_t_hat, const float* __restrict__ Wp,
                         const float* __restrict__ bp, float* __restrict__ dec_fea) {
    int b = blockIdx.x, n = threadIdx.x;
    const float* sb = s_t_hat + b * N2_;
    const float* wr = Wp + (long)n * N2_;
    float acc = bp[n];
    #pragma unroll 4
    for (int k = 0; k < N2_; ++k) acc += sb[k] * wr[k];
    dec_fea[b * N2_ + n] = acc;
}

// ---------------------------------------------------------------------------
// K4: scores[b,t] = sum_n tanh(enc_feat[b,t,n] + dec_fea[b,n] + cov[b,t]*Wc[n]) * v[n]
// grid(B, TK/8), block(32,8): one wave32 per (b,t)
__global__ void k_scores(const float* __restrict__ enc_feat, const float* __restrict__ dec_fea,
                         const float* __restrict__ cov, const float* __restrict__ Wc,
                         const float* __restrict__ v_w, float* __restrict__ scores) {
    int b = blockIdx.x;
    int t = blockIdx.y * 8 + threadIdx.y;
    int lane = threadIdx.x;
    const float4* ef = (const float4*)(enc_feat + ((long)b * TK_ + t) * N2_);
    const float4* df = (const float4*)(dec_fea + b * N2_);
    const float4* wc = (const float4*)Wc;
    const float4* vv = (const float4*)v_w;
    float cv = cov[b * TK_ + t];
    float s = 0.f;
    #pragma unroll
    for (int j = 0; j < 4; ++j) {            // 512 floats = 4 * (32 lanes * 4)
        int idx = j * 32 + lane;             // float4 index
        float4 e4 = ef[idx], d4 = df[idx], w4 = wc[idx], v4 = vv[idx];
        s += tanhf(e4.x + d4.x + cv * w4.x) * v4.x;
        s += tanhf(e4.y + d4.y + cv * w4.y) * v4.y;
        s += tanhf(e4.z + d4.z + cv * w4.z) * v4.z;
        s += tanhf(e4.w + d4.w + cv * w4.w) * v4.w;
    }
    #pragma unroll
    for (int m = 16; m >= 1; m >>= 1) s += __shfl_xor(s, m, 32);
    if (lane == 0) scores[b * TK_ + t] = s;
}

// ---------------------------------------------------------------------------
// K5: attn = softmax(scores)*mask, renormalized; coverage_out = cov + attn
// grid(B), block(256)
__global__ void k_attn(const float* __restrict__ scores, const float* __restrict__ mask,
                       const float* __restrict__ cov_in,
                       float* __restrict__ attn, float* __restrict__ cov_out) {
    __shared__ float red[256];
    int b = blockIdx.x, tid = threadIdx.x;
    const float* sb = scores + b * TK_;
    float m = -1e30f;
    for (int t = tid; t < TK_; t += 256) m = fmaxf(m, sb[t]);
    red[tid] = m; __syncthreads();
    for (int s = 128; s >= 1; s >>= 1) { if (tid < s) red[tid] = fmaxf(red[tid], red[tid + s]); __syncthreads(); }
    m = red[0]; __syncthreads();
    float sum = 0.f;
    for (int t = tid; t < TK_; t += 256) {
        float w = expf(sb[t] - m) * mask[b * TK_ + t];
        attn[b * TK_ + t] = w;
        sum += w;
    }
    red[tid] = sum; __syncthreads();
    for (int s = 128; s >= 1; s >>= 1) { if (tid < s) red[tid] += red[tid + s]; __syncthreads(); }
    float inv = 1.f / red[0];
    for (int t = tid; t < TK_; t += 256) {
        float a = attn[b * TK_ + t] * inv;
        attn[b * TK_ + t] = a;
        cov_out[b * TK_ + t] = cov_in[b * TK_ + t] + a;
    }
}

// ---------------------------------------------------------------------------
// K6: c_t[b,n] = sum_t attn[b,t] * enc_out[b,t,n]
// grid(B), block(512)
__global__ void k_ct(const float* __restrict__ attn, const float* __restrict__ enc_out,
                     float* __restrict__ c_t) {
    __shared__ float aS[TK_];
    int b = blockIdx.x, n = threadIdx.x;
    for (int t = n; t < TK_; t += 512) aS[t] = attn[b * TK_ + t];
    __syncthreads();
    const float* eb = enc_out + (long)b * TK_ * N2_;
    float acc = 0.f;
    for (int t = 0; t < TK_; ++t) {
        __builtin_prefetch(eb + (t + 4) * N2_ + n, 0, 1);   // global_prefetch_b8
        acc += aS[t] * eb[t * N2_ + n];
    }
    c_t[b * N2_ + n] = acc;
}

// ---------------------------------------------------------------------------
// K7: p_gen = sigmoid(concat(c_t, s_t_hat, x) @ Wpg.T + bpg)   [B]
// grid(B), block(128)
__global__ void k_pgen(const float* __restrict__ c_t, const float* __restrict__ s_t_hat,
                       const float* __restrict__ x, const float* __restrict__ Wpg,
                       const float* __restrict__ bpg, float* __restrict__ p_gen) {
    __shared__ float red[128];
    int b = blockIdx.x, tid = threadIdx.x;
    float acc = 0.f;
    for (int k = tid; k < N2_; k += 128) acc += c_t[b * N2_ + k] * Wpg[k];
    for (int k = tid; k < N2_; k += 128) acc += s_t_hat[b * N2_ + k] * Wpg[N2_ + k];
    acc += x[b * EMB_ + tid] * Wpg[2 * N2_ + tid];
    red[tid] = acc; __syncthreads();
    for (int s = 64; s >= 1; s >>= 1) { if (tid < s) red[tid] += red[tid + s]; __syncthreads(); }
    if (tid == 0) p_gen[b] = 1.f / (1.f + expf(-(red[0] + bpg[0])));
}

// ---------------------------------------------------------------------------
// K8: output = concat(h1, c_t) @ W1.T + b1    [B,256]
// grid(B), block(256)
__global__ void k_out1(const float* __restrict__ h1, const float* __restrict__ c_t,
                       const float* __restrict__ W1, const float* __restrict__ b1,
                       float* __restrict__ outp) {
    int b = blockIdx.x, n = threadIdx.x;
    const float* wr = W1 + (long)n * (3 * H_);
    const float* hb = h1 + b * H_;
    const float* cb = c_t + b * N2_;
    float acc = b1[n];
    #pragma unroll 4
    for (int k = 0; k < H_; ++k) acc += hb[k] * wr[k];
    #pragma unroll 4
    for (int k = 0; k < N2_; ++k) acc += cb[k] * wr[H_ + k];
    outp[b * H_ + n] = acc;
}

// ---------------------------------------------------------------------------
// K9: logits = output @ W2.T + b2 via V_WMMA_F32_16X16X4_F32
// Register-blocked: one wave32 computes a 64x64 logits block (MT_ x NT_ tiles):
// 16 independent WMMA chains per k-step fed by 4 A-frag + 4 B-frag b64 loads.
// __launch_bounds__(32,1) lifts the 128-VGPR occupancy cap so the 128
// accumulator VGPRs stay resident (round-2 codegen spilled them to scratch).
// A/B addressed as single base pointer + constant 16KB tile offsets (fits the
// 24-bit instruction offset) — no pointer arrays to spill. Tail handled by
// clamping ntbase (overlapping blocks rewrite identical values: deterministic).
// fp32 A 16x4 layout: lane<16 -> M=lane, K={0,1}; lane>=16 -> M=lane-16, K={2,3}
// fp32 B 4x16 layout: lane<16 -> N=lane, K={0,1}; lane>=16 -> N=lane-16, K={2,3}
// fp32 C/D 16x16:     VGPR v -> M = v + (lane<16 ? 0 : 8), N = lane%16
// grid(ceil(3125/NT_), 2), block(32). Writes into d_out final region (stride VP1_).
__global__ void __launch_bounds__(32, 1)
k_logits_wmma(const float* __restrict__ outp, const float* __restrict__ W2,
              const float* __restrict__ b2, float* __restrict__ logits) {
    int lane = threadIdx.x;
    int half = lane >> 4;          // 0: K={0,1}, 1: K={2,3}
    int lid  = lane & 15;
    int ntbase = blockIdx.x * NT_;
    if (ntbase > NTILES_ - NT_) ntbase = NTILES_ - NT_;
    int m0 = blockIdx.y * (MT_ * 16);

    const float* Abase = outp + (long)(m0 + lid) * H_ + 2 * half;
    const float* Bbase = W2 + (long)(ntbase * 16 + lid) * H_ + 2 * half;

    v8f acc[MT_][NT_] = {};
    for (int k = 0; k < H_; k += 4) {
        v2f a[MT_], bfr[NT_];
        #pragma unroll
        for (int mt = 0; mt < MT_; ++mt)
            a[mt] = *(const v2f*)(Abase + mt * (16 * H_) + k);
        #pragma unroll
        for (int nt = 0; nt < NT_; ++nt)
            bfr[nt] = *(const v2f*)(Bbase + nt * (16 * H_) + k);
        #pragma unroll
        for (int mt = 0; mt < MT_; ++mt) {
            #pragma unroll
            for (int nt = 0; nt < NT_; ++nt) {
                acc[mt][nt] = __builtin_amdgcn_wmma_f32_16x16x4_f32(
                        /*neg_a=*/false, a[mt], /*neg_b=*/false, bfr[nt],
                        /*c_mod=*/(short)0, acc[mt][nt],
                        /*reuse_a=*/false, /*reuse_b=*/false);
            }
        }
    }

    #pragma unroll
    for (int nt = 0; nt < NT_; ++nt) {
        int n0 = (ntbase + nt) * 16;
        float bias = b2[n0 + lid];
        #pragma unroll
        for (int mt = 0; mt < MT_; ++mt) {
            #pragma unroll
            for (int v = 0; v < 8; ++v) {
                int m = m0 + mt * 16 + v + 8 * half;
                logits[(long)m * VP1_ + n0 + lid] = acc[mt][nt][v] + bias;
            }
        }
    }
}

// ---------------------------------------------------------------------------
// K10: in-place over d_out final region: row = p_gen * softmax(row[:V]); tail = extra_zeros
// grid(B), block(1024)
__global__ void k_softmaxV(float* __restrict__ fin, const float* __restrict__ p_gen,
                           const float* __restrict__ extra_zeros) {
    __shared__ float red[1024];
    int b = blockIdx.x, tid = threadIdx.x;
    float* row = fin + (long)b * VP1_;
    float m = -1e30f;
    for (int i = tid; i < V_; i += 1024) m = fmaxf(m, row[i]);
    red[tid] = m; __syncthreads();
    for (int s = 512; s >= 1; s >>= 1) { if (tid < s) red[tid] = fmaxf(red[tid], red[tid + s]); __syncthreads(); }
    m = red[0]; __syncthreads();
    float sum = 0.f;
    for (int i = tid; i < V_; i += 1024) sum += expf(row[i] - m);
    red[tid] = sum; __syncthreads();
    for (int s = 512; s >= 1; s >>= 1) { if (tid < s) red[tid] += red[tid + s]; __syncthreads(); }
    float scale = p_gen[b] / red[0];
    for (int i = tid; i < V_; i += 1024) row[i] = expf(row[i] - m) * scale;
    if (tid < OOV_) row[V_ + tid] = extra_zeros[b * OOV_ + tid];
}

// ---------------------------------------------------------------------------
// K11: scatter-add pointer distribution: fin[b, idx[b,t]] += (1-p_gen[b]) * attn[b,t]
// grid(B), block(512) with guard
__global__ void k_scatter(float* __restrict__ fin, const int* __restrict__ ext_vocab,
                          const float* __restrict__ attn, const float* __restrict__ p_gen) {
    int b = blockIdx.x, t = threadIdx.x;
    if (t >= TK_) return;
    float w = (1.f - p_gen[b]) * attn[b * TK_ + t];
    atomicAdd(fin + (long)b * VP1_ + ext_vocab[b * TK_ + t], w);
}

// ---------------------------------------------------------------------------
extern "C" void kernel_launch(void* const* d_in, const int* in_sizes, int n_in,
                              void* d_out, int out_size, void* d_ws, size_t ws_size,
                              hipStream_t stream) {
    // Inputs in setup_inputs() order
    const int*   y_t_1     = (const int*)  d_in[0];
    const float* s_t_h     = (const float*)d_in[1];   // [1,B,H] -> h0
    const float* s_t_c     = (const float*)d_in[2];   // c0
    const float* enc_out   = (const float*)d_in[3];   // [B,TK,512]
    const float* enc_feat  = (const float*)d_in[4];   // [B*TK,512]
    const float* mask      = (const float*)d_in[5];
    const float* c_t_1     = (const float*)d_in[6];
    const float* extra_z   = (const float*)d_in[7];
    const int*   ext_vocab = (const int*)  d_in[8];
    const float* cov_in    = (const float*)d_in[9];
    // d_in[10] = step (unused: training path taken unconditionally)
    const float* embedding = (const float*)d_in[11];
    const float* Wx  = (const float*)d_in[12];
    const float* bx  = (const float*)d_in[13];
    const float* w_ih = (const float*)d_in[14];
    const float* b_ih = (const float*)d_in[15];
    const float* w_hh = (const float*)d_in[16];
    const float* b_hh = (const float*)d_in[17];
    const float* Wp  = (const float*)d_in[18];
    const float* bp  = (const float*)d_in[19];
    const float* v_w = (const float*)d_in[20];
    const float* Wc  = (const float*)d_in[21];
    const float* Wpg = (const float*)d_in[22];
    const float* bpg = (const float*)d_in[23];
    const float* W1  = (const float*)d_in[24];
    const float* b1  = (const float*)d_in[25];
    const float* W2  = (const float*)d_in[26];
    const float* b2  = (const float*)d_in[27];

    // Output layout (flat, return order)
    float* out        = (float*)d_out;
    float* out_final  = out;                                   // B*(V+OOV)
    float* out_h1     = out_final + (long)B_ * VP1_;           // B*H
    float* out_c1     = out_h1 + B_ * H_;                      // B*H
    float* out_ct     = out_c1 + B_ * H_;                      // B*N2
    float* out_attn   = out_ct + B_ * N2_;                     // B*TK
    float* out_pgen   = out_attn + B_ * TK_;                   // B
    float* out_cov    = out_pgen + B_;                         // B*TK

    // Workspace (small; logits live in d_out final region)
    float* ws       = (float*)d_ws;
    float* ws_x     = ws;                    // B*128
    float* ws_sthat = ws_x + B_ * EMB_;      // B*512
    float* ws_dfea  = ws_sthat + B_ * N2_;   // B*512
    float* ws_score = ws_dfea + B_ * N2_;    // B*400
    float* ws_outp  = ws_score + B_ * TK_;   // B*256

    k_x     <<<B_, 128, 0, stream>>>(c_t_1, y_t_1, embedding, Wx, bx, ws_x);
    k_lstm  <<<B_, 256, 0, stream>>>(ws_x, s_t_h, s_t_c, w_ih, b_ih, w_hh, b_hh,
                                     ws_sthat, out_h1, out_c1);
    k_decfea<<<B_, 512, 0, stream>>>(ws_sthat, Wp, bp, ws_dfea);
    k_scores<<<dim3(B_, TK_ / 8), dim3(32, 8), 0, stream>>>(enc_feat, ws_dfea, cov_in,
                                                            Wc, v_w, ws_score);
    k_attn  <<<B_, 256, 0, stream>>>(ws_score, mask, cov_in, out_attn, out_cov);
    k_ct    <<<B_, 512, 0, stream>>>(out_attn, enc_out, out_ct);
    k_pgen  <<<B_, 128, 0, stream>>>(out_ct, ws_sthat, ws_x, Wpg, bpg, out_pgen);
    k_out1  <<<B_, 256, 0, stream>>>(out_h1, out_ct, W1, b1, ws_outp);
    k_logits_wmma<<<dim3((NTILES_ + NT_ - 1) / NT_, B_ / (MT_ * 16)), 32, 0, stream>>>(
        ws_outp, W2, b2, out_final);
    k_softmaxV   <<<B_, 1024, 0, stream>>>(out_final, out_pgen, extra_z);
    k_scatter    <<<B_, 512, 0, stream>>>(out_final, ext_vocab, out_attn, out_pgen);
}